// ContinuousNormalizingFlow_29222957482274
// MI455X (gfx1250) — compile-verified
//
#include <hip/hip_runtime.h>
#include <math.h>

typedef float v2f __attribute__((ext_vector_type(2)));
typedef float v8f __attribute__((ext_vector_type(8)));

#define SPITCH 68  // staging row pitch in floats (padded vs 64 to kill bank conflicts)

__device__ __forceinline__ v8f wmma_f32(v2f a, v2f b, v8f c) {
  // D = A(16x4 f32) * B(4x16 f32) + C(16x16 f32)
  return __builtin_amdgcn_wmma_f32_16x16x4_f32(false, a, false, b, (short)0, c, false, false);
}

__device__ __forceinline__ float act_tanh(float x) {
#if __has_builtin(__builtin_amdgcn_tanhf)
  return __builtin_amdgcn_tanhf(x);
#elif __has_builtin(__builtin_amdgcn_tanh_f32)
  return __builtin_amdgcn_tanh_f32(x);
#else
  return tanhf(x);
#endif
}

// A-fragment (rows of the wave's 16xK activation tile) from LDS staging.
__device__ __forceinline__ v2f afrag(const float* st, int l16, int half, int k0) {
  return *(const v2f*)(st + l16 * SPITCH + k0 + 2 * half);  // element offset even -> 8B aligned
}
// B-fragment from pre-swizzled weight pairs: W_pk[q][n] = (B[2q][n], B[2q+1][n])
__device__ __forceinline__ v2f bfrag(const v2f* Wpk, int pitch, int k0, int n, int half) {
  return Wpk[((k0 >> 1) + half) * pitch + n];
}
// Store a 16x16 C tile (v8f per lane) into row-major staging at column n0.
__device__ __forceinline__ void storeC(float* st, int l16, int half, int n0, v8f c) {
#pragma unroll
  for (int v = 0; v < 8; ++v) st[(v + 8 * half) * SPITCH + n0 + l16] = c[v];
}

__global__ __launch_bounds__(256) void cnf_fused_kernel(
    const float* __restrict__ x0, const float* __restrict__ W1, const float* __restrict__ b1,
    const float* __restrict__ W2, const float* __restrict__ b2, const float* __restrict__ W3,
    const float* __restrict__ b3, const int* __restrict__ nsp, float* __restrict__ xout,
    float* __restrict__ ldout, int nrows) {
  // ---- weights in LDS, pre-swizzled into WMMA B-operand pair layout ----
  __shared__ v2f sW1f[4 * 64];   // fwd L1: B[k][n] = W1[n][k], k<8
  __shared__ v2f sW1b[32 * 16];  // bwd:    B[h][d] = W1[h][d], d padded to 16 w/ zeros
  __shared__ v2f sW2f[32 * 64];  // fwd L2: B[k][n] = W2[n][k]
  __shared__ v2f sW2b[32 * 64];  // bwd:    B[h][j] = W2[h][j]
  __shared__ v2f sW3p[32 * 16];  // fwd L3: B[h][d] = W3[d][h], d padded to 16 w/ zeros
  __shared__ float sb1[64], sw1t[64], sb2[64], sc3[64], sb3p[16];
  __shared__ float stg[8][16 * SPITCH];  // per-wave activation staging (C-layout <-> A-layout)

  const int tid = threadIdx.x;
  for (int i = tid; i < 4 * 64; i += 256) {
    int q = i >> 6, n = i & 63;
    v2f v; v.x = W1[n * 9 + 2 * q]; v.y = W1[n * 9 + 2 * q + 1];
    sW1f[i] = v;
  }
  for (int i = tid; i < 32 * 16; i += 256) {
    int q = i >> 4, n = i & 15;
    v2f v; v.x = (n < 8) ? W1[(2 * q) * 9 + n] : 0.f; v.y = (n < 8) ? W1[(2 * q + 1) * 9 + n] : 0.f;
    sW1b[i] = v;
    v2f w; w.x = (n < 8) ? W3[n * 64 + 2 * q] : 0.f; w.y = (n < 8) ? W3[n * 64 + 2 * q + 1] : 0.f;
    sW3p[i] = w;
  }
  for (int i = tid; i < 32 * 64; i += 256) {
    int q = i >> 6, n = i & 63;
    v2f a; a.x = W2[n * 64 + 2 * q]; a.y = W2[n * 64 + 2 * q + 1];
    sW2f[i] = a;
    v2f bb; bb.x = W2[(2 * q) * 64 + n]; bb.y = W2[(2 * q + 1) * 64 + n];
    sW2b[i] = bb;
  }
  if (tid < 64) {
    sb1[tid] = b1[tid];
    sw1t[tid] = W1[tid * 9 + 8];  // time column of W1
    sb2[tid] = b2[tid];
    float s = 0.f;
#pragma unroll
    for (int d = 0; d < 8; ++d) s += W3[d * 64 + tid];
    sc3[tid] = s;  // grad_h2 from ones cotangent is this constant vector
  }
  if (tid < 16) sb3p[tid] = (tid < 8) ? b3[tid] : 0.f;
  __syncthreads();

  const int lane = tid & 31, wave = tid >> 5;
  const int half = lane >> 4, l16 = lane & 15;
  float* st = stg[wave];
  const int row = blockIdx.x * 128 + wave * 16 + l16;
  const bool valid = (row < nrows);

  float xr[8];  // full row held per lane (duplicated across halves)
#pragma unroll
  for (int d = 0; d < 8; ++d) xr[d] = valid ? x0[row * 8 + d] : 0.f;
  float ldrow = 0.f;

  const int nsteps = *nsp;
  const float dt = 1.0f / (float)nsteps;

  // layer1: h1 = tanh(x @ W1a^T + (b1 + t*W1[:,8])) ; tiles kept in regs AND staged
  auto fwd_layer1 = [&](float t_, v8f h1o[4]) {
#pragma unroll
    for (int nt = 0; nt < 4; ++nt) {
      const int n = nt * 16 + l16;
      const float be = sb1[n] + t_ * sw1t[n];
      v8f acc;
#pragma unroll
      for (int j = 0; j < 8; ++j) acc[j] = be;
#pragma unroll
      for (int k0 = 0; k0 < 8; k0 += 4) {
        v2f a; a.x = xr[k0 + 2 * half]; a.y = xr[k0 + 2 * half + 1];
        acc = wmma_f32(a, bfrag(sW1f, 64, k0, n, half), acc);
      }
#pragma unroll
      for (int j = 0; j < 8; ++j) acc[j] = act_tanh(acc[j]);
      h1o[nt] = acc;
      storeC(st, l16, half, nt * 16, acc);
    }
  };
  // layer2: h2 = tanh(h1(staged) @ W2^T + b2), result in regs
  auto fwd_layer2 = [&](v8f h2o[4]) {
#pragma unroll
    for (int nt = 0; nt < 4; ++nt) {
      const int n = nt * 16 + l16;
      v8f acc;
#pragma unroll
      for (int j = 0; j < 8; ++j) acc[j] = sb2[n];
#pragma unroll
      for (int k0 = 0; k0 < 64; k0 += 4)
        acc = wmma_f32(afrag(st, l16, half, k0), bfrag(sW2f, 64, k0, n, half), acc);
#pragma unroll
      for (int j = 0; j < 8; ++j) acc[j] = act_tanh(acc[j]);
      h2o[nt] = acc;
    }
  };
  auto store_tiles = [&](v8f tr[4]) {
#pragma unroll
    for (int nt = 0; nt < 4; ++nt) storeC(st, l16, half, nt * 16, tr[nt]);
  };
  // [16x64](staged) @ B[64x16] -> single 16x16 tile
  auto mm_stA_n16 = [&](const v2f* Bw, v8f acc) {
#pragma unroll
    for (int k0 = 0; k0 < 64; k0 += 4)
      acc = wmma_f32(afrag(st, l16, half, k0), bfrag(Bw, 16, k0, l16, half), acc);
    return acc;
  };

#pragma unroll 1
  for (int i = 0; i < nsteps; ++i) {
    const float t = (float)i * dt;
    v8f h1r[4], h2r[4];

    // ===== forward at x -> dx; x += dt*dx =====
    fwd_layer1(t, h1r);
    fwd_layer2(h2r);
    store_tiles(h2r);  // h2 -> staging (A-operand for layer3)
    {
      v8f acc;
#pragma unroll
      for (int j = 0; j < 8; ++j) acc[j] = sb3p[l16];
      acc = mm_stA_n16(sW3p, acc);       // dx (cols 8..15 exactly zero)
      storeC(st, l16, half, 0, acc);     // C-layout -> row-major for the x update
    }
#pragma unroll
    for (int d = 0; d < 8; ++d) xr[d] += dt * st[l16 * SPITCH + d];

    // ===== forward at x_new (keep h1,h2 for tanh') + VJP of ones =====
    fwd_layer1(t, h1r);
    fwd_layer2(h2r);
    // g2 = c3 * (1 - h2^2)
#pragma unroll
    for (int nt = 0; nt < 4; ++nt) {
      const float c3v = sc3[nt * 16 + l16];
#pragma unroll
      for (int j = 0; j < 8; ++j) {
        float v = h2r[nt][j];
        h2r[nt][j] = c3v * (1.f - v * v);
      }
    }
    store_tiles(h2r);  // g2 -> staging
    // g1 = (g2 @ W2) * (1 - h1^2)
#pragma unroll
    for (int nt = 0; nt < 4; ++nt) {
      const int n = nt * 16 + l16;
      v8f acc;
#pragma unroll
      for (int j = 0; j < 8; ++j) acc[j] = 0.f;
#pragma unroll
      for (int k0 = 0; k0 < 64; k0 += 4)
        acc = wmma_f32(afrag(st, l16, half, k0), bfrag(sW2b, 64, k0, n, half), acc);
#pragma unroll
      for (int j = 0; j < 8; ++j) {
        float v = h1r[nt][j];
        h2r[nt][j] = acc[j] * (1.f - v * v);
      }
    }
    store_tiles(h2r);  // g1 -> staging
    // grad_x = g1 @ W1a ; ld += dt * sum_d grad_x
    {
      v8f acc;
#pragma unroll
      for (int j = 0; j < 8; ++j) acc[j] = 0.f;
      acc = mm_stA_n16(sW1b, acc);
      storeC(st, l16, half, 0, acc);
    }
    float s = 0.f;
#pragma unroll
    for (int d = 0; d < 8; ++d) s += st[l16 * SPITCH + d];
    ldrow += dt * s;
  }

  if (valid) {
#pragma unroll
    for (int d = 0; d < 4; ++d) xout[row * 8 + 4 * half + d] = xr[4 * half + d];
    if (half == 0) ldout[row] = ldrow;
  }
}

extern "C" void kernel_launch(void* const* d_in, const int* in_sizes, int n_in, void* d_out,
                              int out_size, void* d_ws, size_t ws_size, hipStream_t stream) {
  const float* x0 = (const float*)d_in[0];
  const float* W1 = (const float*)d_in[1];
  const float* b1 = (const float*)d_in[2];
  const float* W2 = (const float*)d_in[3];
  const float* b2 = (const float*)d_in[4];
  const float* W3 = (const float*)d_in[5];
  const float* b3 = (const float*)d_in[6];
  const int* ns = (const int*)d_in[7];
  const int nrows = in_sizes[0] / 8;
  float* xout = (float*)d_out;
  float* ldout = xout + (size_t)nrows * 8;
  const int blocks = (nrows + 127) / 128;
  hipLaunchKernelGGL(cnf_fused_kernel, dim3(blocks), dim3(256), 0, stream, x0, W1, b1, W2, b2, W3,
                     b3, ns, xout, ldout, nrows);
}